// Blur_6863357739161
// MI455X (gfx1250) — compile-verified
//
#include <hip/hip_runtime.h>
#include <hip/hip_bf16.h>
#include <stdint.h>

// ---------------------------------------------------------------------------
// Depthwise 4x4 binomial blur, pad (1,1,1,1), stride 1.
//   in : (8,64,512,512) f32 -> 512 independent 512x512 images
//   out: (8,64,511,511) f32
// Memory-bound (~1.07 GB @ 23.3 TB/s => ~46us floor). Strategy:
//   - Persistent block per (x-tile, image) column: 16 y-tiles.
//   - TDM tensor_load_to_lds in iterate mode (pitched LDS tiles), DOUBLE
//     BUFFERED: TDM(t+1) issued before s_wait_tensorcnt(1) for tile t, so
//     DMA of the next tile overlaps compute of the current one.
//   - Separable [1,3,3,1]/8 passes from LDS, 8 output rows per thread.
// ---------------------------------------------------------------------------

#define IMG_H 512
#define IMG_W 512
#define OUT_H 511
#define OUT_W 511
#define TW 64            // output tile width
#define TH 32            // output tile height
#define NT (IMG_H / TH)  // 16 y-tiles per column
#define LW (TW + 3)      // 67 = loaded width  (halo 1 left, 2 right)
#define LH (TH + 3)      // 35 = loaded height (halo 1 top, 2 bottom)
#define PITCH LW
#define NTHREADS 256

typedef unsigned int v4u __attribute__((ext_vector_type(4)));
typedef int          v8i __attribute__((ext_vector_type(8)));
typedef int          v4i __attribute__((ext_vector_type(4)));

#ifndef HAVE_TDM
#if defined(__has_builtin)
#if __has_builtin(__builtin_amdgcn_tensor_load_to_lds)
#define HAVE_TDM 1
#else
#define HAVE_TDM 0
#endif
#else
#define HAVE_TDM 0
#endif
#endif

__global__ __launch_bounds__(NTHREADS) void blur4x4_dw_kernel(
    const float* __restrict__ x, float* __restrict__ out) {
  __shared__ float buf[2][LH * PITCH];

  const int gx  = blockIdx.x;  // x-tile: 0..7
  const int n   = blockIdx.y;  // image index (B*C): 0..511
  const int tid = threadIdx.x;

  // Column-constant x clipping.
  const int ox0 = gx * TW;
  const int x0n = ox0 - 1;                               // nominal left edge
  const int xLo = x0n < 0 ? 0 : x0n;
  const int xHi = (x0n + LW) > IMG_W ? IMG_W : (x0n + LW);
  const int VW  = xHi - xLo;                             // valid width
  const int skL = xLo - x0n;                             // 0 or 1

  const float* img = x + (size_t)n * IMG_H * IMG_W;
  float* outImg    = out + (size_t)n * OUT_H * OUT_W;

  // Zero both buffers once: covers the never-written padded columns for the
  // whole column walk, plus the top-halo row of tile 0.
  for (int i = tid; i < 2 * LH * PITCH; i += NTHREADS)
    ((float*)buf)[i] = 0.0f;
  __syncthreads();

#if HAVE_TDM
  // Issue one pitched 2D tile DMA: VH rows of VW floats, LDS pitch = PITCH.
  auto issueTdm = [&](int t, float* dst) {
    const int y0n = t * TH - 1;
    const int yLo = y0n < 0 ? 0 : y0n;
    const int yHi = (y0n + LH) > IMG_H ? IMG_H : (y0n + LH);
    const int VH  = yHi - yLo;
    const int skT = yLo - y0n;

    const uint64_t ga =
        (uint64_t)(uintptr_t)(img + (size_t)yLo * IMG_W + xLo);
    const uint32_t ldsAddr =
        (uint32_t)(uintptr_t)(dst + skT * PITCH + skL);

    const uint32_t tDim0 = 1u << 30;      // tensor dims large: region is
    const uint32_t tDim1 = 1u << 30;      // pre-clamped, no OOB clip wanted
    const uint32_t tileD0 = (uint32_t)VW; // one row per "tile", iterated VH x
    const uint64_t s0 = (uint64_t)IMG_W;
    const uint64_t s1 = (uint64_t)IMG_W * IMG_H;

    v4u g0;
    g0[0] = 1u;                                 // count=1, user descriptor
    g0[1] = ldsAddr;                            // lds_addr (bytes)
    g0[2] = (uint32_t)ga;                       // global_addr[31:0]
    g0[3] = (uint32_t)(ga >> 32) | (2u << 30);  // global_addr hi | type=2

    v8i g1;
    g1[0] = (int)((2u << 16) | (1u << 19));     // data_size=4B, iterate_enable
    g1[1] = (int)((tDim0 & 0xFFFFu) << 16);     // barrier_addr=0 | tdim0.lo
    g1[2] = (int)((tDim0 >> 16) | ((tDim1 & 0xFFFFu) << 16));
    g1[3] = (int)((tDim1 >> 16) | (tileD0 << 16));
    g1[4] = (int)1;                             // tile_dim1=1, tile_dim2=0
    g1[5] = (int)(uint32_t)s0;
    g1[6] = (int)((uint32_t)(s0 >> 32) | (((uint32_t)s1 & 0xFFFFu) << 16));
    g1[7] = (int)(uint32_t)(s1 >> 16);

    v4i g2;
    g2[0] = (int)(1u << 30);                    // tensor_dim2 (unused, large)
    g2[1] = (int)PITCH;                         // lds_addr_increment / iter
    g2[2] = (int)IMG_W;                         // global_addr_increment lo
    g2[3] = (int)(((uint32_t)(VH - 1)) << 16);  // inc.hi=0 | iterate_count
    v4i g3 = {0, 0, 0, 0};

#if __clang_major__ >= 23
    v8i g4 = {0, 0, 0, 0, 0, 0, 0, 0};
    __builtin_amdgcn_tensor_load_to_lds(g0, g1, g2, g3, g4, 0);
#else
    __builtin_amdgcn_tensor_load_to_lds(g0, g1, g2, g3, 0);
#endif
  };

  if (tid < 32) issueTdm(0, buf[0]);  // prologue: tile 0 in flight
#endif

  const float K0 = 0.125f, K1 = 0.375f;  // [1,3,3,1]/8 per axis
  const int tx = tid & (TW - 1);         // 0..63: output column in tile
  const int tg = tid >> 6;               // 0..3 : row-group (8 rows each)
  const int r0 = tg * 8;
  const int ox = ox0 + tx;

  for (int t = 0; t < NT; ++t) {
#if HAVE_TDM
    // Bottom tile loads only 33 of 35 rows; re-zero its 2 stale rows in the
    // target buffer before its TDM is issued (buffer is idle here).
    if (t + 1 == NT - 1) {
      float* nb = buf[(t + 1) & 1];
      for (int i = tid; i < 2 * PITCH; i += NTHREADS)
        nb[33 * PITCH + i] = 0.0f;
      __syncthreads();
    }
    if (tid < 32) {
      if (t + 1 < NT) {
        issueTdm(t + 1, buf[(t + 1) & 1]);  // overlap next DMA with compute
#if defined(__has_builtin) && __has_builtin(__builtin_amdgcn_s_wait_tensorcnt)
        __builtin_amdgcn_s_wait_tensorcnt(1);  // tile t done, t+1 in flight
#else
        asm volatile("s_wait_tensorcnt 0x1" ::: "memory");
#endif
      } else {
#if defined(__has_builtin) && __has_builtin(__builtin_amdgcn_s_wait_tensorcnt)
        __builtin_amdgcn_s_wait_tensorcnt(0);
#else
        asm volatile("s_wait_tensorcnt 0x0" ::: "memory");
#endif
      }
    }
    __syncthreads();  // tile t's LDS data visible to all waves
#else
    // Fallback: synchronous cooperative staging of tile t.
    {
      const int y0n = t * TH - 1;
      float* dst = buf[t & 1];
      for (int i = tid; i < LH * PITCH; i += NTHREADS) {
        const int r = i / PITCH, c = i % PITCH;
        const int iy = y0n + r, ix = x0n + c;
        float v = 0.0f;
        if ((unsigned)iy < (unsigned)IMG_H && (unsigned)ix < (unsigned)IMG_W)
          v = img[(size_t)iy * IMG_W + ix];
        dst[i] = v;
      }
    }
    __syncthreads();
#endif

    // ---- separable blur of tile t from LDS ----
    const float* tb = buf[t & 1];
    float h[11];  // horizontal pass: 11 filtered rows feed 8 outputs
#pragma unroll
    for (int r = 0; r < 11; ++r) {
      const float* p = &tb[(r0 + r) * PITCH + tx];
      h[r] = K0 * (p[0] + p[3]) + K1 * (p[1] + p[2]);
    }
#pragma unroll
    for (int k = 0; k < 8; ++k) {
      const float v = K0 * (h[k] + h[k + 3]) + K1 * (h[k + 1] + h[k + 2]);
      const int oy = t * TH + r0 + k;
      if (oy < OUT_H && ox < OUT_W)
        outImg[(size_t)oy * OUT_W + ox] = v;
    }
    __syncthreads();  // all reads of buf[t&1] done before TDM(t+2) reuses it
  }
}

extern "C" void kernel_launch(void* const* d_in, const int* in_sizes, int n_in,
                              void* d_out, int out_size, void* d_ws,
                              size_t ws_size, hipStream_t stream) {
  (void)in_sizes; (void)n_in; (void)out_size; (void)d_ws; (void)ws_size;
  const float* x = (const float*)d_in[0];
  float* out = (float*)d_out;
  dim3 grid((OUT_W + TW - 1) / TW,  // 8 x-tiles
            8 * 64,                 // B*C images
            1);
  blur4x4_dw_kernel<<<grid, NTHREADS, 0, stream>>>(x, out);
}